// H2HGCN_28836410425411
// MI455X (gfx1250) — compile-verified
//
#include <hip/hip_runtime.h>
#include <math.h>

#define NN   100000
#define EE   1600000
#define DIMM 65
#define TILE 256

static constexpr float EPSF       = 1e-6f;
static constexpr float SELU_SCALE = 1.0507009873554805f;
static constexpr float SELU_ALPHA = 1.6732632423543772f;

#if __has_builtin(__builtin_amdgcn_global_load_async_to_lds_b32) && \
    __has_builtin(__builtin_amdgcn_s_wait_asynccnt)
#define H2H_HAS_ASYNC 1
#else
#define H2H_HAS_ASYNC 0
#endif

// Casts for the async global->LDS builtin: param0 is AS(1) int*, param1 AS(3) int*.
#define H2H_GPTR(p) ((__attribute__((address_space(1))) int*)(p))
#define H2H_LPTR(p) ((__attribute__((address_space(3))) int*)(p))

// wave32 butterfly sum (gfx1250 is wave32-only)
__device__ __forceinline__ float wave_sum32(float v) {
    v += __shfl_xor(v, 16, 32);
    v += __shfl_xor(v, 8, 32);
    v += __shfl_xor(v, 4, 32);
    v += __shfl_xor(v, 2, 32);
    v += __shfl_xor(v, 1, 32);
    return v;
}

// ---------------------------------------------------------------------------
// Edge aggregation (hot kernel, placed first so its asm shows in the snippet).
// Each block stages TILE edges (row,col,w) into LDS via CDNA5 async
// global->LDS loads, then 8 waves each process 32 edges:
//   w2 = w * lamb[col];  num[row] += w2 * k[col];  deg[row] += w2
// Lane j owns feature elements 2j,2j+1: one float2 (b64) gather per edge,
// two f32 atomics. k/num/deg all live in the 192MB L2 -> L2-atomic bound.
// ---------------------------------------------------------------------------
__global__ void h2h_edge_agg(const int* __restrict__ rows,
                             const int* __restrict__ cols,
                             const float* __restrict__ w,
                             const float* __restrict__ k,
                             const float* __restrict__ lamb,
                             float* __restrict__ num,
                             float* __restrict__ deg) {
    __shared__ int   s_row[TILE];
    __shared__ int   s_col[TILE];
    __shared__ float s_w[TILE];

    int tid  = threadIdx.x;
    int base = blockIdx.x * TILE;
    int idx  = base + tid;

#if H2H_HAS_ASYNC
    if (idx < EE) {
        __builtin_amdgcn_global_load_async_to_lds_b32(
            H2H_GPTR(rows + idx), H2H_LPTR(s_row + tid), 0, 0);
        __builtin_amdgcn_global_load_async_to_lds_b32(
            H2H_GPTR(cols + idx), H2H_LPTR(s_col + tid), 0, 0);
        __builtin_amdgcn_global_load_async_to_lds_b32(
            H2H_GPTR(w + idx),    H2H_LPTR(s_w + tid),   0, 0);
    }
    __builtin_amdgcn_s_wait_asynccnt(0);
    __syncthreads();
#else
    if (idx < EE) {
        s_row[tid] = rows[idx];
        s_col[tid] = cols[idx];
        s_w[tid]   = w[idx];
    }
    __syncthreads();
#endif

    int lane = tid & 31;
    int wave = tid >> 5;  // 0..7

    int nE = EE - base;
    if (nE > TILE) nE = TILE;

    for (int i = wave; i < nE; i += 8) {
        int   c  = s_col[i];   // wave-uniform LDS broadcast
        int   r  = s_row[i];
        float w2 = s_w[i] * lamb[c];

        // Prefetch the k-row of an edge ~4 iterations ahead (global_prefetch_b8)
        int ip = i + 32;
        if (ip < nE) __builtin_prefetch(&k[(size_t)s_col[ip] * 64], 0, 0);

        // lane j gathers elements 2j, 2j+1 of the 64-float row: one b64 load
        const float2* krow = (const float2*)(k + (size_t)c * 64);
        float2 kc = krow[lane];

        size_t ro = (size_t)r * 64 + 2 * lane;
        atomicAdd(&num[ro],     w2 * kc.x);
        atomicAdd(&num[ro + 1], w2 * kc.y);
        if (lane == 0) atomicAdd(&deg[r], w2);
    }
}

// ---------------------------------------------------------------------------
// Kernel A: per node (one wave32 per node): k = x[1:]/x[0], lamb = lorenz(k),
// and zero the accumulators for this node. Scalar loads: the 65-float row
// pitch breaks 8B alignment, but lanes are contiguous so fully coalesced.
// ---------------------------------------------------------------------------
__global__ void h2h_node_prep(const float* __restrict__ x,
                              float* __restrict__ k,
                              float* __restrict__ lamb,
                              float* __restrict__ num,
                              float* __restrict__ deg) {
    int node = (blockIdx.x * blockDim.x + threadIdx.x) >> 5;
    int lane = threadIdx.x & 31;
    if (node >= NN) return;

    const float* xr = x + (size_t)node * DIMM;
    float t  = xr[0];                 // time component, >= 1
    float a  = xr[1 + lane];
    float b  = xr[33 + lane];
    float k0 = a / t;
    float k1 = b / t;

    float n2 = wave_sum32(k0 * k0 + k1 * k1);
    n2 = fminf(n2, 0.9f);             // clip(., 0, 0.9); sum >= 0 already
    float g = 1.0f / sqrtf(1.0f - n2);

    size_t ko = (size_t)node * 64;
    k[ko + lane]        = k0;
    k[ko + 32 + lane]   = k1;
    num[ko + lane]      = 0.0f;
    num[ko + 32 + lane] = 0.0f;
    if (lane == 0) {
        lamb[node] = g;
        deg[node]  = 0.0f;
    }
}

// ---------------------------------------------------------------------------
// Kernel C: per node (one wave32 per node): divide by deg, then fused
// k2h -> lorentz_to_poincare -> selu -> poincare_to_lorentz -> normalize.
// ---------------------------------------------------------------------------
__global__ void h2h_node_finish(const float* __restrict__ num,
                                const float* __restrict__ deg,
                                float* __restrict__ out) {
    int node = (blockIdx.x * blockDim.x + threadIdx.x) >> 5;
    int lane = threadIdx.x & 31;
    if (node >= NN) return;

    size_t no = (size_t)node * 64;
    float  d    = deg[node];
    float  dinv = (d > 0.0f) ? (1.0f / d) : 0.0f;
    float  km0  = num[no + lane] * dinv;
    float  km1  = num[no + 32 + lane] * dinv;

    // k2h: g = 1/sqrt(1 - clip(|k|^2, 0, 0.9)); x = [g, g*k]
    float n2 = fminf(wave_sum32(km0 * km0 + km1 * km1), 0.9f);
    float g  = 1.0f / sqrtf(1.0f - n2);

    // lorentz_to_poincare: p = (g*k) / (g + 1)
    float ginv = 1.0f / (g + 1.0f);
    float p0   = g * km0 * ginv;
    float p1   = g * km1 * ginv;

    // selu
    float s0 = (p0 > 0.0f) ? SELU_SCALE * p0
                           : SELU_SCALE * SELU_ALPHA * (expf(p0) - 1.0f);
    float s1 = (p1 > 0.0f) ? SELU_SCALE * p1
                           : SELU_SCALE * SELU_ALPHA * (expf(p1) - 1.0f);

    // poincare_to_lorentz: y = [1+sq, 2s] / (1 - sq + eps), then
    // normalize: sp = y[1:], t = sqrt(1 + |sp|^2)
    float sq    = wave_sum32(s0 * s0 + s1 * s1);
    float inv   = 1.0f / (1.0f - sq + EPSF);
    float sp0   = 2.0f * s0 * inv;
    float sp1   = 2.0f * s1 * inv;
    float spsq  = 4.0f * sq * inv * inv;
    float tcomp = sqrtf(1.0f + spsq);

    float* orow = out + (size_t)node * DIMM;
    orow[1 + lane]  = sp0;
    orow[33 + lane] = sp1;
    if (lane == 0) orow[0] = tcomp;
}

// ---------------------------------------------------------------------------
extern "C" void kernel_launch(void* const* d_in, const int* in_sizes, int n_in,
                              void* d_out, int out_size, void* d_ws, size_t ws_size,
                              hipStream_t stream) {
    (void)in_sizes; (void)n_in; (void)out_size; (void)ws_size;

    const float* node_repr = (const float*)d_in[0];
    const int*   eidx      = (const int*)d_in[1];
    const float* ew        = (const float*)d_in[2];
    const int*   rows      = eidx;        // edge_index[0]
    const int*   cols      = eidx + EE;   // edge_index[1]

    float* ws   = (float*)d_ws;
    float* k    = ws;                       // N*64
    float* lamb = ws + (size_t)NN * 64;     // N
    float* num  = ws + (size_t)NN * 65;     // N*64
    float* deg  = ws + (size_t)NN * 129;    // N
    float* xt   = ws + (size_t)NN * 130;    // N*65 (intermediate layer output)
    float* outp = (float*)d_out;

    const int blk   = 256;                      // 8 wave32 per block
    const int gridN = (NN + 7) / 8;             // one wave per node
    const int gridE = (EE + TILE - 1) / TILE;   // one tile per block

    // Layer 1
    h2h_node_prep<<<gridN, blk, 0, stream>>>(node_repr, k, lamb, num, deg);
    h2h_edge_agg<<<gridE, blk, 0, stream>>>(rows, cols, ew, k, lamb, num, deg);
    h2h_node_finish<<<gridN, blk, 0, stream>>>(num, deg, xt);

    // Layer 2
    h2h_node_prep<<<gridN, blk, 0, stream>>>(xt, k, lamb, num, deg);
    h2h_edge_agg<<<gridE, blk, 0, stream>>>(rows, cols, ew, k, lamb, num, deg);
    h2h_node_finish<<<gridN, blk, 0, stream>>>(num, deg, outp);
}